// TrajPreAttnAvgLongUser2_90211493085342
// MI455X (gfx1250) — compile-verified
//
#include <hip/hip_runtime.h>
#include <hip/hip_bf16.h>
#include <math.h>

typedef __attribute__((ext_vector_type(2))) float v2f;
typedef __attribute__((ext_vector_type(8))) float v8f;

#define SEQ  1024   // target_len == SEQ
#define DX   544    // LOC_EMB + TIM_EMB
#define DZ   608    // DX + UID_EMB
#define NLOC 50000
#define LOCE 512
#define TIME 32
#define UIDE 64

// ---------------------------------------------------------------------------
// Gather x = [emb_loc[loc] | emb_tim[tim]]  -> X (SEQ x DX)
// ---------------------------------------------------------------------------
__global__ void __launch_bounds__(256)
k_gather_x(const int* __restrict__ loc, const int* __restrict__ tim,
           const float* __restrict__ eloc, const float* __restrict__ etim,
           float* __restrict__ X) {
    int i  = blockIdx.x;
    int li = loc[i], ti = tim[i];
    for (int j = threadIdx.x; j < DX; j += 256) {
        float v = (j < LOCE) ? eloc[(size_t)li * LOCE + j]
                             : etim[(size_t)ti * TIME + (j - LOCE)];
        X[(size_t)i * DX + j] = v;
    }
}

// ---------------------------------------------------------------------------
// Generic f32 WMMA GEMM: C(1024 x N) = act(A(1024 x K) @ B + bias)
// TRANS_B: B given as X[n][k] row-major (i.e. B = X^T logically)
// EPI: 0 = none, 1 = sigmoid, 2 = tanh  (bias added when non-null)
// Block = 256 threads = 8 waves, wave grid 4(M) x 2(N); block tile 64 x 128.
// Wave tile 16 x 64 via 4x V_WMMA_F32_16X16X4_F32 accumulators.
//
// K-loop: 2-deep software pipeline, 8-wide K step. Fragments consumed in
// iteration i were loaded in the same prefetch slot of iteration i-1, i.e.
// load->use distance is ~10 outstanding b64 loads, so the compiler can keep
// s_wait_loadcnt well above zero and overlap VMEM latency with WMMA issue.
// Requires K % 8 == 0 (holds: 544, 608, 1024).
// ---------------------------------------------------------------------------
template<bool TRANS_B, int EPI>
__global__ void __launch_bounds__(256)
k_wmma_gemm(const float* __restrict__ A, int lda,
            const float* __restrict__ B, int ldb,
            float* __restrict__ C, int ldc,
            int N, int K, const float* __restrict__ bias) {
    const int lane = threadIdx.x & 31;
    const int wave = threadIdx.x >> 5;
    const int wm   = wave & 3;
    const int wn   = wave >> 2;
    const int half = lane >> 4;   // 0: lanes 0-15, 1: lanes 16-31
    const int lsub = lane & 15;

    const int m0 = blockIdx.y * 64 + wm * 16;
    const int n0 = blockIdx.x * 128 + wn * 64;

    v8f acc[4] = {};

    // A fragment base: lane<16 -> (M=m0+lsub, K={k0,k0+1}); lane>=16 -> K={k0+2,k0+3}
    const float* aptr = A + (size_t)(m0 + lsub) * lda + 2 * half;

    int coln[4];
    const float* bptr[4];
#pragma unroll
    for (int t = 0; t < 4; ++t) {
        int col  = n0 + t * 16 + lsub;
        coln[t]  = col;
        int c2   = col < N ? col : N - 1;   // clamp; store is guarded
        bptr[t]  = TRANS_B ? (B + (size_t)c2 * ldb + 2 * half) : (B + c2);
    }

    auto loadB = [&](int kk, int t) -> v2f {
        v2f b;
        if (TRANS_B) {
            b = *(const v2f*)(bptr[t] + kk);
        } else {
            const float* bp = bptr[t] + (size_t)(kk + 2 * half) * ldb;
            b.x = bp[0];
            b.y = bp[ldb];
        }
        return b;
    };

    // ---- 2-deep pipeline prologue: stage0 = k 0, stage1 = k 4 ----
    v2f a0 = *(const v2f*)(aptr);
    v2f a1 = *(const v2f*)(aptr + 4);
    v2f b0[4], b1[4];
#pragma unroll
    for (int t = 0; t < 4; ++t) b0[t] = loadB(0, t);
#pragma unroll
    for (int t = 0; t < 4; ++t) b1[t] = loadB(4, t);

    for (int k0 = 0; k0 < K; k0 += 8) {
        // clamped prefetch indices (branch-free; redundant in-bounds reload at tail)
        const int kp0 = (k0 + 8  < K) ? (k0 + 8)  : k0;
        const int kp1 = (k0 + 12 < K) ? (k0 + 12) : k0;

        if (!TRANS_B) {
            // pull the strided weight stream further ahead (global_prefetch_b8)
            const int kpp = (k0 + 24 < K) ? (k0 + 24) : k0;
            __builtin_prefetch(bptr[0] + (size_t)(kpp + 2 * half) * ldb, 0, 1);
        }

        // issue loads for k0+8 while computing k0
        v2f a0n = *(const v2f*)(aptr + kp0);
        v2f b0n[4];
#pragma unroll
        for (int t = 0; t < 4; ++t) b0n[t] = loadB(kp0, t);

#pragma unroll
        for (int t = 0; t < 4; ++t) {
            acc[t] = __builtin_amdgcn_wmma_f32_16x16x4_f32(
                false, a0, false, b0[t], (short)0, acc[t], false, false);
        }

        // issue loads for k0+12 while computing k0+4
        v2f a1n = *(const v2f*)(aptr + kp1);
        v2f b1n[4];
#pragma unroll
        for (int t = 0; t < 4; ++t) b1n[t] = loadB(kp1, t);

#pragma unroll
        for (int t = 0; t < 4; ++t) {
            acc[t] = __builtin_amdgcn_wmma_f32_16x16x4_f32(
                false, a1, false, b1[t], (short)0, acc[t], false, false);
        }

        a0 = a0n;
        a1 = a1n;
#pragma unroll
        for (int t = 0; t < 4; ++t) { b0[t] = b0n[t]; b1[t] = b1n[t]; }
    }

#pragma unroll
    for (int t = 0; t < 4; ++t) {
        if (coln[t] >= N) continue;
        float bv = bias ? bias[coln[t]] : 0.0f;
#pragma unroll
        for (int v = 0; v < 8; ++v) {
            float val = acc[t][v] + bv;
            if (EPI == 1) val = 1.0f / (1.0f + __expf(-val));
            if (EPI == 2) val = tanhf(val);
            int row = m0 + 8 * half + v;
            C[(size_t)row * ldc + coln[t]] = val;
        }
    }
}

// ---------------------------------------------------------------------------
// Row softmax (in place), row length SEQ
// ---------------------------------------------------------------------------
__global__ void __launch_bounds__(256)
k_softmax_rows(float* __restrict__ S) {
    __shared__ float redm[8];
    __shared__ float reds[8];
    int tid = threadIdx.x;
    float* p = S + (size_t)blockIdx.x * SEQ;

    float m = -3.4e38f;
    for (int j = tid; j < SEQ; j += 256) m = fmaxf(m, p[j]);
    for (int off = 16; off; off >>= 1) m = fmaxf(m, __shfl_xor(m, off, 32));
    if ((tid & 31) == 0) redm[tid >> 5] = m;
    __syncthreads();
    m = redm[0];
#pragma unroll
    for (int w = 1; w < 8; ++w) m = fmaxf(m, redm[w]);

    float s = 0.0f;
    for (int j = tid; j < SEQ; j += 256) {
        float e = __expf(p[j] - m);
        p[j] = e;
        s += e;
    }
    for (int off = 16; off; off >>= 1) s += __shfl_xor(s, off, 32);
    if ((tid & 31) == 0) reds[tid >> 5] = s;
    __syncthreads();
    s = 0.0f;
#pragma unroll
    for (int w = 0; w < 8; ++w) s += reds[w];
    float inv = 1.0f / s;
    for (int j = tid; j < SEQ; j += 256) p[j] *= inv;
}

// ---------------------------------------------------------------------------
// Elementwise helpers
// ---------------------------------------------------------------------------
__global__ void __launch_bounds__(256)
k_mul(const float* __restrict__ a, const float* __restrict__ b,
      float* __restrict__ c, int n) {
    int i = blockIdx.x * 256 + threadIdx.x;
    if (i < n) c[i] = a[i] * b[i];
}

// ZC(row, :544) = h + (1-z)*(g-h);  ZC(row, 544:608) = emb_uid[uid]
__global__ void __launch_bounds__(256)
k_combine(const float* __restrict__ H, const float* __restrict__ Z,
          const float* __restrict__ G, const float* __restrict__ euid,
          const int* __restrict__ uid, float* __restrict__ ZC) {
    int i = blockIdx.x;
    int u = uid[0];
    for (int j = threadIdx.x; j < DZ; j += 256) {
        float v;
        if (j < DX) {
            size_t id = (size_t)i * DX + j;
            float h = H[id];
            v = h + (1.0f - Z[id]) * (G[id] - h);
        } else {
            v = euid[(size_t)u * UIDE + (j - DX)];
        }
        ZC[(size_t)i * DZ + j] = v;
    }
}

// Per-row  m + log(sum exp(y - m))  over NLOC columns
__global__ void __launch_bounds__(256)
k_rowlse(const float* __restrict__ Y, float* __restrict__ rowc) {
    __shared__ float redm[8];
    __shared__ float reds[8];
    int tid = threadIdx.x;
    const float* p = Y + (size_t)blockIdx.x * NLOC;

    float m = -3.4e38f;
    for (int j = tid; j < NLOC; j += 256) m = fmaxf(m, p[j]);
    for (int off = 16; off; off >>= 1) m = fmaxf(m, __shfl_xor(m, off, 32));
    if ((tid & 31) == 0) redm[tid >> 5] = m;
    __syncthreads();
    m = redm[0];
#pragma unroll
    for (int w = 1; w < 8; ++w) m = fmaxf(m, redm[w]);

    float s = 0.0f;
    for (int j = tid; j < NLOC; j += 256) s += __expf(p[j] - m);
    for (int off = 16; off; off >>= 1) s += __shfl_xor(s, off, 32);
    if ((tid & 31) == 0) reds[tid >> 5] = s;
    __syncthreads();
    s = 0.0f;
#pragma unroll
    for (int w = 0; w < 8; ++w) s += reds[w];
    if (tid == 0) rowc[blockIdx.x] = m + __logf(s);
}

__global__ void __launch_bounds__(256)
k_sub_rowc(float* __restrict__ Y, const float* __restrict__ rowc) {
    int row = blockIdx.y;
    int j = blockIdx.x * 256 + threadIdx.x;
    if (j < NLOC) Y[(size_t)row * NLOC + j] -= rowc[row];
}

// ---------------------------------------------------------------------------
extern "C" void kernel_launch(void* const* d_in, const int* in_sizes, int n_in,
                              void* d_out, int out_size, void* d_ws, size_t ws_size,
                              hipStream_t stream) {
    (void)in_sizes; (void)n_in; (void)out_size; (void)ws_size;
    const int*   loc   = (const int*)d_in[0];
    const int*   tim   = (const int*)d_in[1];
    // d_in[2..4]: history_* — provably dead w.r.t. the output (pointwise GRU)
    const int*   uid   = (const int*)d_in[5];
    // d_in[6]: target_len == SEQ (compile-time constant here)
    const float* eloc  = (const float*)d_in[7];
    const float* etim  = (const float*)d_in[8];
    const float* euid  = (const float*)d_in[9];
    const float* gru_w = (const float*)d_in[10];
    const float* gru_b = (const float*)d_in[11];
    const float* fcw   = (const float*)d_in[12];
    const float* fcb   = (const float*)d_in[13];
    float* out = (float*)d_out;

    float* ws   = (float*)d_ws;
    float* X    = ws;                 // SEQ*DX
    float* S    = X  + SEQ * DX;      // SEQ*SEQ
    float* XA   = S  + SEQ * SEQ;     // SEQ*DX
    float* R    = XA + SEQ * DX;      // SEQ*DX
    float* Zg   = R  + SEQ * DX;      // SEQ*DX
    float* RH   = Zg + SEQ * DX;      // SEQ*DX
    float* G    = RH + SEQ * DX;      // SEQ*DX
    float* ZC   = G  + SEQ * DX;      // SEQ*DZ
    float* rowc = ZC + SEQ * DZ;      // SEQ

    // 1) embedding gather
    k_gather_x<<<SEQ, 256, 0, stream>>>(loc, tim, eloc, etim, X);

    // 2) S = X @ X^T  (TRANS_B: B rows are X rows)
    dim3 g1((SEQ + 127) / 128, SEQ / 64);
    k_wmma_gemm<true, 0><<<g1, 256, 0, stream>>>(X, DX, X, DX, S, SEQ, SEQ, DX, nullptr);

    // 3) row softmax in place
    k_softmax_rows<<<SEQ, 256, 0, stream>>>(S);

    // 4) XA = S @ X
    dim3 g2((DX + 127) / 128, SEQ / 64);
    k_wmma_gemm<false, 0><<<g2, 256, 0, stream>>>(S, SEQ, X, DX, XA, DX, DX, SEQ, nullptr);

    // 5) r = sigmoid(XA @ W0 + b0); z = sigmoid(XA @ W1 + b1)
    k_wmma_gemm<false, 1><<<g2, 256, 0, stream>>>(XA, DX, gru_w,           DX, R,  DX, DX, DX, gru_b);
    k_wmma_gemm<false, 1><<<g2, 256, 0, stream>>>(XA, DX, gru_w + DX * DX, DX, Zg, DX, DX, DX, gru_b + DX);

    // 6) rh = r * h ; g = tanh(rh @ W2 + b2)
    k_mul<<<(SEQ * DX + 255) / 256, 256, 0, stream>>>(R, XA, RH, SEQ * DX);
    k_wmma_gemm<false, 2><<<g2, 256, 0, stream>>>(RH, DX, gru_w + 2 * DX * DX, DX, G, DX, DX, DX, gru_b + 2 * DX);

    // 7) out_state + concat uid embedding -> ZC (SEQ x DZ)
    k_combine<<<SEQ, 256, 0, stream>>>(XA, Zg, G, euid, uid, ZC);

    // 8) Y = ZC @ fc_final_w + fc_final_b  -> d_out (SEQ x NLOC)
    dim3 g3((NLOC + 127) / 128, SEQ / 64);
    k_wmma_gemm<false, 0><<<g3, 256, 0, stream>>>(ZC, DZ, fcw, NLOC, out, NLOC, NLOC, DZ, fcb);

    // 9) log_softmax: Y -= (max + log sum exp) per row
    k_rowlse<<<SEQ, 256, 0, stream>>>(out, rowc);
    dim3 g4((NLOC + 255) / 256, SEQ);
    k_sub_rowc<<<g4, 256, 0, stream>>>(out, rowc);
}